// CenterLoss_40673340293427
// MI455X (gfx1250) — compile-verified
//
#include <hip/hip_runtime.h>
#include <stdint.h>

#define N_FEATURE 256
#define N_CLASS   100000
#define BATCH     262144
#define ALPHA     0.1f

#define ROWS_PER_WAVE   4
#define WAVES_PER_BLOCK 8
#define ROWS_PER_BLOCK  (ROWS_PER_WAVE * WAVES_PER_BLOCK)   // 32
#define TILE_ELEMS      (ROWS_PER_WAVE * N_FEATURE)         // 1024 f32 = 4 KB / wave

typedef uint32_t u32;
typedef uint64_t u64;
typedef u32 u32x4 __attribute__((ext_vector_type(4)));
typedef u32 u32x8 __attribute__((ext_vector_type(8)));

// ---------------------------------------------------------------------------
// Kernel 1: zero the scatter accumulators (must run every call: atomics below).
// `vbase` is guaranteed 16B-aligned by the host; nvec float4 stores + tail.
// ---------------------------------------------------------------------------
__global__ void cl_init_kernel(float4* __restrict__ vbase, u32 nvec, u32 ntail,
                               u32* __restrict__ counts, double* loss_acc) {
    u32 i      = blockIdx.x * blockDim.x + threadIdx.x;
    u32 stride = gridDim.x * blockDim.x;
    const float4 z4 = make_float4(0.f, 0.f, 0.f, 0.f);
    for (u32 t = i; t < nvec; t += stride) vbase[t] = z4;          // global_store_b128
    if (i < ntail) ((float*)(vbase + nvec))[i] = 0.0f;
    for (u32 t = i; t < (u32)N_CLASS; t += stride) counts[t] = 0u;
    if (i == 0) *loss_acc = 0.0;
}

// ---------------------------------------------------------------------------
// Kernel 2: per-wave TDM staging of h rows into LDS, gather center, scatter
//           sums/counts with global atomics, f64 atomic loss accumulation.
// ---------------------------------------------------------------------------
__global__ __launch_bounds__(256) void
cl_main_kernel(const float* __restrict__ h, const int* __restrict__ dcls,
               const float* __restrict__ center, float* __restrict__ sums,
               u32* __restrict__ counts, double* __restrict__ loss_acc)
{
    __shared__ float sh[ROWS_PER_BLOCK * N_FEATURE];      // 32 KB

    const int lane = threadIdx.x & 31;
    const int wave = threadIdx.x >> 5;
    const int r0   = blockIdx.x * ROWS_PER_BLOCK + wave * ROWS_PER_WAVE;

    // ---- async stage: TDM-copy this wave's 4 contiguous h rows (4 KB) into
    //      its private LDS slice. D# per CDNA5 ISA.
    {
        u64 gaddr = (u64)(uintptr_t)(h + (size_t)r0 * N_FEATURE);
        u32 laddr = (u32)(uintptr_t)(&sh[wave * TILE_ELEMS]);  // LDS byte offset
        u32x4 g0;
        g0[0] = 1u;                                   // count=1, user descriptor
        g0[1] = laddr;                                // lds_addr[31:0]
        g0[2] = (u32)gaddr;                           // global_addr[31:0]
        g0[3] = (u32)(gaddr >> 32) | (2u << 30);      // global_addr[56:32] | type=2
        u32x8 g1;
        g1[0] = 2u << 16;                             // wg_mask=0 (not in cluster), data_size=4B
        g1[1] = ((u32)TILE_ELEMS & 0xFFFFu) << 16;    // atomic_bar=0 | tensor_dim0[15:0]
        g1[2] = ((u32)TILE_ELEMS >> 16) | (1u << 16); // tensor_dim0[31:16] | tensor_dim1[15:0]=1
        g1[3] = (u32)TILE_ELEMS << 16;                // tensor_dim1[31:16]=0 | tile_dim0=1024
        g1[4] = 1u;                                   // tile_dim1=1 | tile_dim2=0
        g1[5] = (u32)TILE_ELEMS;                      // tensor_dim0_stride[31:0]
        g1[6] = ((u32)TILE_ELEMS & 0xFFFFu) << 16;    // stride0[47:32]=0 | stride1[15:0]
        g1[7] = 0u;                                   // tensor_dim1_stride[47:16]
        asm volatile("tensor_load_to_lds %0, %1" :: "s"(g0), "s"(g1) : "memory");
    }

    // ---- overlap the DMA: fetch class ids and prefetch the gathered
    //      center rows toward the caches (global_prefetch_b8).
    int cls[ROWS_PER_WAVE];
#pragma unroll
    for (int r = 0; r < ROWS_PER_WAVE; ++r) {
        cls[r] = dcls[r0 + r];
        __builtin_prefetch(&center[(size_t)cls[r] * N_FEATURE + lane * 8], 0, 3);
    }

    asm volatile("s_wait_tensorcnt 0x0" ::: "memory");   // tile resident in LDS

    float acc = 0.0f;
    const float* slice = &sh[wave * TILE_ELEMS];
#pragma unroll
    for (int r = 0; r < ROWS_PER_WAVE; ++r) {
        const int c = cls[r];
        const float4* hp = (const float4*)(slice  + r * N_FEATURE + lane * 8);   // ds_load_b128
        const float4* cp = (const float4*)(center + (size_t)c * N_FEATURE + lane * 8);
        float4 h0 = hp[0], h1 = hp[1];
        float4 c0 = cp[0], c1 = cp[1];
        float t;
        t = h0.x - c0.x; acc = fmaf(t, t, acc);
        t = h0.y - c0.y; acc = fmaf(t, t, acc);
        t = h0.z - c0.z; acc = fmaf(t, t, acc);
        t = h0.w - c0.w; acc = fmaf(t, t, acc);
        t = h1.x - c1.x; acc = fmaf(t, t, acc);
        t = h1.y - c1.y; acc = fmaf(t, t, acc);
        t = h1.z - c1.z; acc = fmaf(t, t, acc);
        t = h1.w - c1.w; acc = fmaf(t, t, acc);
        float* sp = sums + (size_t)c * N_FEATURE + lane * 8;   // global_atomic_add_f32 ×8
        atomicAdd(sp + 0, h0.x); atomicAdd(sp + 1, h0.y);
        atomicAdd(sp + 2, h0.z); atomicAdd(sp + 3, h0.w);
        atomicAdd(sp + 4, h1.x); atomicAdd(sp + 5, h1.y);
        atomicAdd(sp + 6, h1.z); atomicAdd(sp + 7, h1.w);
        if (lane == 0) atomicAdd(counts + c, 1u);
    }

    // wave32 butterfly reduction of the squared-diff partial, one f64 atomic/wave
#pragma unroll
    for (int off = 16; off > 0; off >>= 1)
        acc += __shfl_down(acc, off, 32);
    if (lane == 0) atomicAdd(loss_acc, (double)acc);
}

// ---------------------------------------------------------------------------
// Kernel 3: new_center = center + ALPHA*(sums/counts - center) where present;
//           out[0] = loss_acc / (B*F).  4 elems/thread; NT loads of the dead
//           sums buffer, NT stores of the streamed output (out+1 is only
//           4B-aligned, so stores stay b32). sums may alias out+1 -> no
//           restrict, read-before-write within the owning thread only.
// ---------------------------------------------------------------------------
__global__ void cl_finalize_kernel(const float* __restrict__ center,
                                   const float* sums,
                                   const u32* __restrict__ counts,
                                   const double* __restrict__ loss_acc,
                                   float* out)
{
    u32 i      = blockIdx.x * blockDim.x + threadIdx.x;
    u32 stride = gridDim.x * blockDim.x;
    const u32 NV = (u32)((size_t)N_CLASS * N_FEATURE / 4);   // 6.4M float4 slots
    for (u32 v = i; v < NV; v += stride) {
        u32    k   = v >> 6;                       // (4v)/256: class of this slot
        u32    cnt = counts[k];
        float4 c4  = *(const float4*)(center + (size_t)v * 4);
        float  s0  = __builtin_nontemporal_load(sums + (size_t)v * 4 + 0);
        float  s1  = __builtin_nontemporal_load(sums + (size_t)v * 4 + 1);
        float  s2  = __builtin_nontemporal_load(sums + (size_t)v * 4 + 2);
        float  s3  = __builtin_nontemporal_load(sums + (size_t)v * 4 + 3);
        float  r0 = c4.x, r1 = c4.y, r2 = c4.z, r3 = c4.w;
        if (cnt > 0u) {
            float inv = 1.0f / (float)cnt;
            r0 = fmaf(ALPHA, fmaf(s0, inv, -c4.x), c4.x);
            r1 = fmaf(ALPHA, fmaf(s1, inv, -c4.y), c4.y);
            r2 = fmaf(ALPHA, fmaf(s2, inv, -c4.z), c4.z);
            r3 = fmaf(ALPHA, fmaf(s3, inv, -c4.w), c4.w);
        }
        float* o = out + 1 + (size_t)v * 4;
        __builtin_nontemporal_store(r0, o + 0);
        __builtin_nontemporal_store(r1, o + 1);
        __builtin_nontemporal_store(r2, o + 2);
        __builtin_nontemporal_store(r3, o + 3);
    }
    if (i == 0)
        out[0] = (float)(*loss_acc / (double)((size_t)BATCH * N_FEATURE));
}

// ---------------------------------------------------------------------------
extern "C" void kernel_launch(void* const* d_in, const int* in_sizes, int n_in,
                              void* d_out, int out_size, void* d_ws, size_t ws_size,
                              hipStream_t stream)
{
    (void)in_sizes; (void)n_in; (void)out_size;
    const float* h      = (const float*)d_in[0];
    const int*   dcls   = (const int*)d_in[1];
    const float* center = (const float*)d_in[2];
    float*       out    = (float*)d_out;

    // workspace layout: [0,16): f64 loss accumulator; [16, ...): u32 counts;
    // then (if it fits) a 256B-aligned f32 sums buffer, else reuse out+1 in place.
    double* loss_acc = (double*)d_ws;
    u32*    counts   = (u32*)((char*)d_ws + 16);
    size_t  sums_off = (16 + sizeof(u32) * (size_t)N_CLASS + 255) & ~(size_t)255;
    const size_t NS  = (size_t)N_CLASS * N_FEATURE;

    float* sums;
    float4* zero_base;          // 16B-aligned region to clear
    u32 nvec, ntail;
    if (ws_size >= sums_off + sizeof(float) * NS) {
        sums      = (float*)((char*)d_ws + sums_off);
        zero_base = (float4*)sums;                   // 256B-aligned
        nvec      = (u32)(NS / 4);  ntail = 0;
    } else {
        sums      = out + 1;        // only 4-byte accesses ever touch sums -> safe
        zero_base = (float4*)out;   // clear loss slot + sums together, aligned
        nvec      = (u32)((NS + 1) / 4);  ntail = (u32)((NS + 1) % 4);
    }

    cl_init_kernel<<<4096, 256, 0, stream>>>(zero_base, nvec, ntail, counts, loss_acc);
    cl_main_kernel<<<BATCH / ROWS_PER_BLOCK, 256, 0, stream>>>(h, dcls, center,
                                                               sums, counts, loss_acc);
    cl_finalize_kernel<<<8192, 256, 0, stream>>>(center, sums, counts, loss_acc, out);
}